// RateRNN_91036126806978
// MI455X (gfx1250) — compile-verified
//
#include <hip/hip_runtime.h>

#define N_   1024
#define T_   256
#define B_   4
#define BT_  1024                 // B_*T_
#define BNT  (B_ * N_ * T_)       // 1048576
#define S_   100
#define DT_  0.0005f

typedef __attribute__((ext_vector_type(2))) float v2f;
typedef __attribute__((ext_vector_type(8))) float v8f;

// One Euler step: rec = W @ relu(u_prev), fused pointwise update, writes history slot.
//
// LDS tile layout is pair-interleaved: element (m, c) of the 64x64 relu(u) chunk
// lives at lds_r[(m>>1)*128 + 2*c + (m&1)], so a WMMA B-fragment (rows kk,kk+1 of
// column c, kk even) is one contiguous 8-byte ds_load_b64.
__global__ __launch_bounds__(128)
void rnn_step(const float* __restrict__ W,
              const float* __restrict__ u_prev,
              const float* __restrict__ eta,
              const float* __restrict__ h,
              const float* __restrict__ bias,
              float* __restrict__ u_out)
{
    __shared__ float lds_r[64 * 64];

    const int tid   = threadIdx.x;
    const int wave  = tid >> 5;
    const int lane  = tid & 31;
    const int lhalf = lane >> 4;   // 0/1 : selects K pair (A/B) or M+8 (C)
    const int lmod  = lane & 15;

    const int n_base = blockIdx.y * 64 + wave * 16;  // output rows (n)
    const int c_base = blockIdx.x * 64;              // flat (b,t) column base
    const int batch  = c_base >> 8;                  // 64 | 256 -> single batch per block
    const int t0     = c_base & 255;

    const float* Wrow  = W + (size_t)(n_base + lmod) * N_;           // A row for this lane
    const float* ubase = u_prev + (size_t)batch * (N_ * T_) + t0;    // B tile source
    const v2f*   lds2  = reinterpret_cast<const v2f*>(lds_r);        // pair view

    v8f acc[4] = {};

    for (int kc = 0; kc < N_; kc += 64) {
        __syncthreads();
        // Cooperative fill: rows m = kc..kc+63 x 64 cols. float4 global loads
        // (coalesced along t), relu, pair-interleaved LDS stores.
        #pragma unroll
        for (int i = 0; i < 8; ++i) {
            const int e   = tid + i * 128;      // 0..1023 float4 elements
            const int row = e >> 4;             // 16 float4 per 64-col row
            const int c4  = (e & 15) << 2;      // starting column
            float4 v = *reinterpret_cast<const float4*>(
                &ubase[(size_t)(kc + row) * T_ + c4]);
            const int base = (row >> 1) * 128 + c4 * 2 + (row & 1);
            lds_r[base    ] = fmaxf(v.x, 0.0f);
            lds_r[base + 2] = fmaxf(v.y, 0.0f);
            lds_r[base + 4] = fmaxf(v.z, 0.0f);
            lds_r[base + 6] = fmaxf(v.w, 0.0f);
        }
        __syncthreads();

        #pragma unroll
        for (int k0 = 0; k0 < 64; k0 += 4) {
            const int kk = k0 + 2 * lhalf;      // even: this lane's K pair
            v2f a = *reinterpret_cast<const v2f*>(&Wrow[kc + kk]);  // global_load_b64
            const int prow = (kk >> 1) * 64;
            #pragma unroll
            for (int j = 0; j < 4; ++j) {
                v2f bf = lds2[prow + j * 16 + lmod];                // ds_load_b64
                acc[j] = __builtin_amdgcn_wmma_f32_16x16x4_f32(
                    /*neg_a=*/false, a, /*neg_b=*/false, bf,
                    /*c_mod=*/(short)0, acc[j], /*reuse_a=*/false, /*reuse_b=*/false);
            }
        }
    }

    // Fused pointwise Euler update + clip, using the 16x16 f32 C/D layout:
    // VGPR v holds M = v + 8*lhalf, lane%16 = N (column).
    #pragma unroll
    for (int j = 0; j < 4; ++j) {
        const int c = c_base + j * 16 + lmod;
        const int t = c & 255;
        #pragma unroll
        for (int v = 0; v < 8; ++v) {
            const int n = n_base + v + 8 * lhalf;
            const size_t idx = (size_t)batch * (N_ * T_) + (size_t)n * T_ + t;
            const float up   = u_prev[idx];
            const float et   = eta[idx];
            const float fv   = fmaxf(h[batch * N_ + n], 0.0f);
            const float bs   = bias[n];
            const float tinv = (n < 512) ? 50.0f  : 100.0f;   // 1/0.02, 1/0.01
            const float umin = (n < 512) ? -100.0f : -80.0f;
            const float umax = (n < 512) ?  100.0f :  80.0f;
            const float rec  = acc[j][v];
            float un = up + DT_ * tinv * (-up + fv + rec + bs + et);
            un = fminf(fmaxf(un, umin), umax);
            u_out[idx] = un;
        }
    }
}

// eta_final = noise[S-1]  (float4 copy, 1M elements)
__global__ __launch_bounds__(256)
void copy4(const float4* __restrict__ src, float4* __restrict__ dst)
{
    int i = blockIdx.x * blockDim.x + threadIdx.x;
    dst[i] = src[i];
}

extern "C" void kernel_launch(void* const* d_in, const int* in_sizes, int n_in,
                              void* d_out, int out_size, void* d_ws, size_t ws_size,
                              hipStream_t stream)
{
    const float* u0    = (const float*)d_in[0];
    const float* h     = (const float*)d_in[1];
    const float* eta0  = (const float*)d_in[2];
    const float* noise = (const float*)d_in[3];
    const float* W     = (const float*)d_in[4];
    const float* bias  = (const float*)d_in[5];
    // d_in[6] = num_steps (device scalar) — statically S_=100 in the reference.

    float* out = (float*)d_out;

    dim3 grid(BT_ / 64, N_ / 64);   // (16, 16)
    for (int s = 0; s < S_; ++s) {
        const float* up = (s == 0) ? u0   : out + (size_t)(s - 1) * BNT;
        const float* et = (s == 0) ? eta0 : noise + (size_t)(s - 1) * BNT;
        rnn_step<<<grid, 128, 0, stream>>>(W, up, et, h, bias, out + (size_t)s * BNT);
    }

    // eta_final appended after the S*B*N*T history
    copy4<<<(BNT / 4) / 256, 256, 0, stream>>>(
        (const float4*)(noise + (size_t)(S_ - 1) * BNT),
        (float4*)(out + (size_t)S_ * BNT));
}